// VectorQuantizer_88115549045196
// MI455X (gfx1250) — compile-verified
//
#include <hip/hip_runtime.h>
#include <hip/hip_bf16.h>

typedef __attribute__((ext_vector_type(2))) float v2f;
typedef __attribute__((ext_vector_type(8))) float v8f;
typedef __attribute__((ext_vector_type(4))) int   v4i;

#define N_ROWS   16384      // 16*32*32
#define D_DIM    64
#define K_CODES  8192
#define KSPLIT   4          // split-K over codebook: 128 x 4 = 512 workgroups
#define K_PER_WG (K_CODES / KSPLIT)       // 2048 codes per WG
#define TILE_K   128        // codes staged in LDS per buffer
#define LDS_PAD  68         // padded row stride (floats): conflict-free, 16B aligned
#define N_TILES  (K_PER_WG / TILE_K)      // 16 tiles per WG
#define Q_OFF    1                         // d_out: [loss][quantized 1048576][indices 16384]
#define I_OFF    (1 + N_ROWS * D_DIM)
#define LOSS_SCALE (1.25f / 1048576.0f)

// workspace layout (floats): [wnorm 8192][pval 16384*4][pidx 16384*4 (as int)]
#define WS_WNORM 0
#define WS_PVAL  (K_CODES)
#define WS_PIDX  (K_CODES + N_ROWS * KSPLIT)

#if defined(__has_builtin)
#if __has_builtin(__builtin_amdgcn_global_load_async_to_lds_b128)
#define USE_ASYNC_LDS 1
#endif
#endif

#ifdef USE_ASYNC_LDS
typedef __attribute__((address_space(1))) v4i gv4i_t;   // global v4i
typedef __attribute__((address_space(3))) v4i lv4i_t;   // LDS v4i
#endif

static __device__ __forceinline__ void wait_async0() {
#ifdef USE_ASYNC_LDS
#if __has_builtin(__builtin_amdgcn_s_wait_asynccnt)
    __builtin_amdgcn_s_wait_asynccnt(0);
#else
    asm volatile("s_wait_asynccnt 0x0" ::: "memory");
#endif
#endif
}

// ---------------------------------------------------------------------------
// Kernel 0: codebook squared norms -> ws  (8192 floats)
// ---------------------------------------------------------------------------
__global__ void vq_wnorm_kernel(const float* __restrict__ W, float* __restrict__ wnorm) {
    int c = blockIdx.x * blockDim.x + threadIdx.x;   // 0..8191
    const float4* w4 = (const float4*)(W + (size_t)c * D_DIM);
    float s = 0.f;
#pragma unroll
    for (int i = 0; i < 16; ++i) {
        float4 v = w4[i];
        s += v.x * v.x + v.y * v.y + v.z * v.z + v.w * v.w;
    }
    wnorm[c] = s;
}

// ---------------------------------------------------------------------------
// Kernel 1: fused distance-GEMM (f32 WMMA) + per-WG argmin partials.
// Grid (128 M-tiles, 4 K-splits); block = 256 threads (8 waves, 16 rows each).
// Double-buffered LDS tiles; next tile streamed via async global->LDS copies
// (ASYNCcnt) while the current tile's 128 WMMAs execute.
// ---------------------------------------------------------------------------
__global__ void __launch_bounds__(256)
vq_main_kernel(const float* __restrict__ X,      // [16384, 64]
               const float* __restrict__ W,      // [8192, 64]
               const float* __restrict__ wnorm,  // [8192]
               float* __restrict__ pval,         // [16384, KSPLIT]
               int*   __restrict__ pidx) {       // [16384, KSPLIT]
    __shared__ float wtile[2][TILE_K * LDS_PAD];  // 2 x 34 KB

    const int lane  = threadIdx.x & 31;
    const int wave  = threadIdx.x >> 5;
    const int mbase = blockIdx.x * 128 + wave * 16;
    const int kbase = blockIdx.y * K_PER_WG;
    const int hlane = lane & 15;          // position within 16-lane half
    const int koff  = (lane >> 4) * 2;    // upper half holds K=2,3 of each 4-chunk

    // per-thread staging geometry: 8 float4 chunks per tile
    const int sc = threadIdx.x >> 4;      // base code row handled (advances by 16)
    const int sq = threadIdx.x & 15;      // float4 slot within the 64-float code row

    // stage one 128x64 tile into LDS buffer `buf` (async if available)
    auto stage = [&](int buf, int wt) {
        const float* gsrc = W + (size_t)(kbase + wt * TILE_K) * D_DIM;
#pragma unroll
        for (int i = 0; i < 8; ++i) {
            const int c = sc + i * 16;                    // code row 0..127
            float*       dst = &wtile[buf][c * LDS_PAD + sq * 4];
            const float* src = gsrc + (size_t)c * D_DIM + sq * 4;
#ifdef USE_ASYNC_LDS
            __builtin_amdgcn_global_load_async_to_lds_b128(
                (gv4i_t*)src, (lv4i_t*)dst, 0, 0);
#else
            *(float4*)dst = *(const float4*)src;          // sync fallback
#endif
        }
    };

    // ---- A operand: 16x64 fp32 tile, register-resident for entire kernel ----
    // WMMA 32-bit A 16x4: lane<16 -> (M=lane, K=k0,k0+1); lane>=16 -> (M=lane-16, K=k0+2,k0+3)
    v2f areg[16];
    {
        const float* arow = X + (size_t)(mbase + hlane) * D_DIM + koff;
#pragma unroll
        for (int kk = 0; kk < 16; ++kk) {
            float2 a2 = *(const float2*)(arow + 4 * kk);
            areg[kk].x = a2.x; areg[kk].y = a2.y;
        }
    }

    // running per-lane argmin: rows (r, r+8) live in lanes (0-15, 16-31), slot r
    float bestv[8];
    int   besti[8];
#pragma unroll
    for (int r = 0; r < 8; ++r) { bestv[r] = 3.4e38f; besti[r] = 0; }

    stage(0, 0);   // prologue: fetch first tile

    for (int wt = 0; wt < N_TILES; ++wt) {
        wait_async0();         // this wave's async copies (tile wt) have landed
        __syncthreads();       // all waves' copies landed; prev buffer free

        if (wt + 1 < N_TILES) {
            stage((wt + 1) & 1, wt + 1);        // overlap fetch of next tile
            if (wt + 2 < N_TILES)               // nudge tile wt+2 toward L2
                __builtin_prefetch(W + (size_t)(kbase + (wt + 2) * TILE_K) * D_DIM, 0, 3);
        }

        const float* bbase = &wtile[wt & 1][hlane * LDS_PAD + koff];

        // ---- 8 column groups of 16 codes each: 16 WMMAs per group ----
#pragma unroll
        for (int cg = 0; cg < 8; ++cg) {
            const float* brow = bbase + cg * 16 * LDS_PAD;

            v8f acc0 = {}; v8f acc1 = {};   // two independent split-K chains
#pragma unroll
            for (int kk = 0; kk < 8; ++kk) {
                float2 b0 = *(const float2*)(brow + 4 * kk);
                float2 b1 = *(const float2*)(brow + 4 * (kk + 8));
                v2f bb0; bb0.x = b0.x; bb0.y = b0.y;
                v2f bb1; bb1.x = b1.x; bb1.y = b1.y;
                acc0 = __builtin_amdgcn_wmma_f32_16x16x4_f32(
                    false, areg[kk],     false, bb0, (short)0, acc0, false, false);
                acc1 = __builtin_amdgcn_wmma_f32_16x16x4_f32(
                    false, areg[kk + 8], false, bb1, (short)0, acc1, false, false);
            }

            const int   ncol = kbase + wt * TILE_K + cg * 16 + hlane;
            const float wn   = wnorm[ncol];
#pragma unroll
            for (int r = 0; r < 8; ++r) {
                float d = wn - 2.0f * (acc0[r] + acc1[r]); // ||x||^2 const per row: omitted
                if (d < bestv[r]) { bestv[r] = d; besti[r] = ncol; }
            }
        }
    }

    // ---- butterfly argmin across each 16-lane half (wave32) ----
#pragma unroll
    for (int m = 1; m < 16; m <<= 1) {
#pragma unroll
        for (int r = 0; r < 8; ++r) {
            float ov = __shfl_xor(bestv[r], m);
            int   oi = __shfl_xor(besti[r], m);
            if (ov < bestv[r] || (ov == bestv[r] && oi < besti[r])) {
                bestv[r] = ov; besti[r] = oi;
            }
        }
    }
    // after butterfly: every lane in a half holds all 8 of its half's row results.
    // lane 0 writes rows mbase..+7, lane 16 writes rows mbase+8..+15 (static indices).
    if (hlane == 0) {
        const int rb = mbase + (lane >> 4) * 8;
#pragma unroll
        for (int r = 0; r < 8; ++r) {
            pval[(size_t)(rb + r) * KSPLIT + blockIdx.y] = bestv[r];
            pidx[(size_t)(rb + r) * KSPLIT + blockIdx.y] = besti[r];
        }
    }
}

// ---------------------------------------------------------------------------
// Kernel 2: merge split-K partials, gather codebook row, write outputs + loss.
// One thread per input row.
// ---------------------------------------------------------------------------
__global__ void __launch_bounds__(256)
vq_finish_kernel(const float* __restrict__ X, const float* __restrict__ W,
                 const float* __restrict__ pval, const int* __restrict__ pidx,
                 float* __restrict__ out) {
    const int row = blockIdx.x * blockDim.x + threadIdx.x;   // 0..16383
    float bv = 3.4e38f; int bi = 0;
#pragma unroll
    for (int k = 0; k < KSPLIT; ++k) {
        float v = pval[(size_t)row * KSPLIT + k];
        int   i = pidx[(size_t)row * KSPLIT + k];
        if (v < bv || (v == bv && i < bi)) { bv = v; bi = i; }
    }
    out[I_OFF + row] = (float)bi;

    const float4* wq = (const float4*)(W + (size_t)bi * D_DIM);
    const float4* xr = (const float4*)(X + (size_t)row * D_DIM);
    float4* qo = (float4*)(out + Q_OFF + (size_t)row * D_DIM);
    float ls = 0.f;
#pragma unroll
    for (int q = 0; q < 16; ++q) {
        float4 w4 = wq[q], x4 = xr[q];
        qo[q] = w4;
        float dx = w4.x - x4.x, dy = w4.y - x4.y;
        float dz = w4.z - x4.z, dw = w4.w - x4.w;
        ls += dx * dx + dy * dy + dz * dz + dw * dw;
    }
#pragma unroll
    for (int m = 1; m < 32; m <<= 1) ls += __shfl_xor(ls, m);
    if ((threadIdx.x & 31) == 0)
        atomicAdd(out, ls * LOSS_SCALE);   // loss = 1.25 * mean((q - x)^2)
}

// ---------------------------------------------------------------------------
extern "C" void kernel_launch(void* const* d_in, const int* in_sizes, int n_in,
                              void* d_out, int out_size, void* d_ws, size_t ws_size,
                              hipStream_t stream) {
    const float* X = (const float*)d_in[0];   // inputs [16,32,32,64]
    const float* W = (const float*)d_in[1];   // codebook [8192,64]
    float* out   = (float*)d_out;
    float* ws    = (float*)d_ws;
    float* wnorm = ws + WS_WNORM;
    float* pval  = ws + WS_PVAL;
    int*   pidx  = (int*)(ws + WS_PIDX);

    // zero the loss accumulator (stream op: graph-capture safe)
    (void)hipMemsetAsync(d_out, 0, sizeof(float), stream);

    vq_wnorm_kernel<<<K_CODES / 256, 256, 0, stream>>>(W, wnorm);
    vq_main_kernel<<<dim3(N_ROWS / 128, KSPLIT), 256, 0, stream>>>(X, W, wnorm, pval, pidx);
    vq_finish_kernel<<<N_ROWS / 256, 256, 0, stream>>>(X, W, pval, pidx, out);
}